// ParT_LHAASO_4561255268384
// MI455X (gfx1250) — compile-verified
//
#include <hip/hip_runtime.h>
#include <hip/hip_bf16.h>
#include <math.h>

typedef __attribute__((ext_vector_type(2))) float v2f;
typedef __attribute__((ext_vector_type(8))) float v8f;

#define Bv 8
#define Nv 384
#define Ev 128
#define Hv 8
#define HD 16

__device__ __forceinline__ float gelu_exact(float x) {
    return 0.5f * x * (1.0f + erff(x * 0.70710678118654752440f));
}

__device__ __forceinline__ v8f wmma4(v2f a, v2f b, v8f c) {
    return __builtin_amdgcn_wmma_f32_16x16x4_f32(
        /*neg_a=*/false, a, /*neg_b=*/false, b,
        /*c_mod=*/(short)0, c, /*reuse_a=*/false, /*reuse_b=*/false);
}

// ---------------------------------------------------------------------------
// WMMA f32 GEMM, register-blocked: one wave -> 16(M) x 64(N) of C.
// 4 accumulators share each A fragment -> 4 back-to-back WMMAs per K-step.
// A frag (16x4): lane%16 = M, lane/16 picks K pair {0,1}/{2,3}
// B frag (4x16): lane%16 = N, lane/16 picks K pair {0,1}/{2,3}
// C frag: VGPR r -> M = r + 8*(lane/16), N = lane%16
// ---------------------------------------------------------------------------
__global__ void wmma_gemm_kernel(const float* __restrict__ A,
                                 const float* __restrict__ W,
                                 const float* __restrict__ bias,
                                 const float* __restrict__ res,
                                 float* __restrict__ C,
                                 int M, int K, int Nc, int act) {
    int wave = (blockIdx.x * blockDim.x + threadIdx.x) >> 5;
    int lane = threadIdx.x & 31;
    int tilesN = Nc >> 6;              // 64-wide N tiles
    int tilesM = M >> 4;
    if (wave >= tilesM * tilesN) return;
    int tm = wave / tilesN;
    int tn = wave - tm * tilesN;

    int mn   = lane & 15;
    int koff = (lane >> 4) << 1;       // 0 or 2

    const float* Arow = A + (size_t)(tm * 16 + mn) * K;
    const float* Wb   = W + tn * 64 + mn;

    v8f acc0 = {}, acc1 = {}, acc2 = {}, acc3 = {};
    for (int k = 0; k < K; k += 4) {
        v2f a;
        a[0] = Arow[k + koff];
        a[1] = Arow[k + koff + 1];
        const float* w0 = Wb + (size_t)(k + koff) * Nc;
        const float* w1 = Wb + (size_t)(k + koff + 1) * Nc;
        v2f b0, b1, b2, b3;
        b0[0] = w0[0];  b0[1] = w1[0];
        b1[0] = w0[16]; b1[1] = w1[16];
        b2[0] = w0[32]; b2[1] = w1[32];
        b3[0] = w0[48]; b3[1] = w1[48];
        acc0 = wmma4(a, b0, acc0);
        acc1 = wmma4(a, b1, acc1);
        acc2 = wmma4(a, b2, acc2);
        acc3 = wmma4(a, b3, acc3);
    }

    int rbase = tm * 16 + ((lane >> 4) << 3);
#pragma unroll
    for (int t = 0; t < 4; ++t) {
        v8f acc = (t == 0) ? acc0 : (t == 1) ? acc1 : (t == 2) ? acc2 : acc3;
        int col = tn * 64 + t * 16 + (lane & 15);
        float bv = bias ? bias[col] : 0.0f;
#pragma unroll
        for (int r = 0; r < 8; ++r) {
            float v = acc[r] + bv;
            if (res) v += res[(size_t)(rbase + r) * Nc + col];
            if (act == 1) v = gelu_exact(v);
            C[(size_t)(rbase + r) * Nc + col] = v;
        }
    }
}

// ---------------------------------------------------------------------------
// LayerNorm over last dim (128), one block (128 threads) per row
// ---------------------------------------------------------------------------
__global__ void ln_kernel(const float* __restrict__ in, float* __restrict__ out,
                          const float* __restrict__ g, const float* __restrict__ b) {
    int row = blockIdx.x;
    int e = threadIdx.x;
    float x = in[(size_t)row * Ev + e];
    __shared__ float s1[4], s2[4];
    float sum = x, sq = x * x;
#pragma unroll
    for (int o = 16; o >= 1; o >>= 1) {
        sum += __shfl_xor(sum, o);
        sq  += __shfl_xor(sq, o);
    }
    if ((e & 31) == 0) { s1[e >> 5] = sum; s2[e >> 5] = sq; }
    __syncthreads();
    float ts = s1[0] + s1[1] + s1[2] + s1[3];
    float tq = s2[0] + s2[1] + s2[2] + s2[3];
    float m = ts * (1.0f / Ev);
    float var = tq * (1.0f / Ev) - m * m;
    out[(size_t)row * Ev + e] = (x - m) * rsqrtf(var + 1e-5f) * g[e] + b[e];
}

// ---------------------------------------------------------------------------
// Embedding: feat = gelu(LN(hits@emb_w+emb_b, g, bn)) + gelu(pos@pos_w+pos_b)
// ---------------------------------------------------------------------------
__global__ void embed_kernel(const float* __restrict__ hits,
                             const float* __restrict__ emb_w, const float* __restrict__ emb_b,
                             const float* __restrict__ emb_g, const float* __restrict__ emb_bn,
                             const float* __restrict__ pos_w, const float* __restrict__ pos_b,
                             float* __restrict__ feat) {
    int row = blockIdx.x;          // b*N+n
    int e = threadIdx.x;           // 0..127
    const float* hp = hits + (size_t)row * 5;
    float t = emb_b[e];
#pragma unroll
    for (int c = 0; c < 5; ++c) t = fmaf(hp[c], emb_w[c * Ev + e], t);
    __shared__ float s1[4], s2[4];
    float sum = t, sq = t * t;
#pragma unroll
    for (int o = 16; o >= 1; o >>= 1) {
        sum += __shfl_xor(sum, o);
        sq  += __shfl_xor(sq, o);
    }
    if ((e & 31) == 0) { s1[e >> 5] = sum; s2[e >> 5] = sq; }
    __syncthreads();
    float m = (s1[0] + s1[1] + s1[2] + s1[3]) * (1.0f / Ev);
    float var = (s2[0] + s2[1] + s2[2] + s2[3]) * (1.0f / Ev) - m * m;
    float x = (t - m) * rsqrtf(var + 1e-5f) * emb_g[e] + emb_bn[e];
    float p = fmaf(hp[0], pos_w[e], fmaf(hp[1], pos_w[Ev + e], pos_b[e]));
    feat[(size_t)row * Ev + e] = gelu_exact(x) + gelu_exact(p);
}

// ---------------------------------------------------------------------------
// denom[b,c] = mean_{i,j} |a_i - a_j| + 1e-6,  a = hits[b,:,2+c]
// ---------------------------------------------------------------------------
__global__ void pair_denom_kernel(const float* __restrict__ hits, float* __restrict__ denom) {
    int bc = blockIdx.x;           // B*3
    int b = bc / 3, c = bc % 3;
    int tid = threadIdx.x;         // 256
    __shared__ float a[Nv];
    for (int n = tid; n < Nv; n += 256) a[n] = hits[(size_t)(b * Nv + n) * 5 + 2 + c];
    __syncthreads();
    float sum = 0.0f;
    for (int idx = tid; idx < Nv * Nv; idx += 256) {
        int i = idx / Nv, j = idx - i * Nv;
        sum += fabsf(a[i] - a[j]);
    }
    __shared__ float red[256];
    red[tid] = sum;
    __syncthreads();
    for (int s = 128; s > 0; s >>= 1) {
        if (tid < s) red[tid] += red[tid + s];
        __syncthreads();
    }
    if (tid == 0) denom[bc] = red[0] / ((float)Nv * (float)Nv) + 1e-6f;
}

// ---------------------------------------------------------------------------
// Pair bias (per layer): bias[b,h,i,j] = relu(d@pw1+pb1)@pw2 + pb2
// One wave -> 16 consecutive (b,i,j) rows. Stage 1 (K=3) in VALU -> LDS
// (row stride 130 to dodge 64-bank conflicts), stage 2 (K=128 -> 8 heads)
// as 32x V_WMMA_F32_16X16X4_F32 against zero-padded pw2 [128x16] in LDS.
// ---------------------------------------------------------------------------
#define PB_WAVES 4
__global__ void pair_bias_kernel(const float* __restrict__ hits, const float* __restrict__ denom,
                                 const float* __restrict__ pw1, const float* __restrict__ pb1,
                                 const float* __restrict__ pw2, const float* __restrict__ pb2,
                                 float* __restrict__ bias) {
    __shared__ float w1s[3 * Ev];
    __shared__ float b1s[Ev];
    __shared__ float w2p[Ev * 16];               // padded [e][16], h<8 valid
    __shared__ float ts[PB_WAVES][16 * 130];     // stage-1 activations
    int tid = threadIdx.x;                       // 128 threads, 4 waves
    for (int k = tid; k < 3 * Ev; k += 128) w1s[k] = pw1[k];
    for (int k = tid; k < Ev; k += 128) b1s[k] = pb1[k];
    for (int k = tid; k < Ev * 16; k += 128) {
        int e = k >> 4, h = k & 15;
        w2p[k] = (h < 8) ? pw2[e * 8 + h] : 0.0f;
    }
    __syncthreads();

    int wave = tid >> 5, lane = tid & 31;
    size_t idx0 = ((size_t)blockIdx.x * PB_WAVES + wave) * 16;   // first row of tile
    int j0 = (int)(idx0 % Nv);                   // tile never crosses i boundary
    size_t r0 = idx0 / Nv;
    int i = (int)(r0 % Nv);
    int b = (int)(r0 / Nv);

    const float* hi = hits + (size_t)(b * Nv + i) * 5;
    float hi0 = hi[2], hi1 = hi[3], hi2 = hi[4];
    float dn0 = denom[b * 3 + 0], dn1 = denom[b * 3 + 1], dn2 = denom[b * 3 + 2];

    for (int r = 0; r < 16; ++r) {
        const float* hj = hits + (size_t)(b * Nv + j0 + r) * 5;
        float d0 = (hi0 - hj[2]) / dn0;
        float d1 = (hi1 - hj[3]) / dn1;
        float d2 = (hi2 - hj[4]) / dn2;
#pragma unroll
        for (int ee = 0; ee < 4; ++ee) {
            int e = lane + 32 * ee;
            float t = fmaf(d0, w1s[e], fmaf(d1, w1s[Ev + e], fmaf(d2, w1s[2 * Ev + e], b1s[e])));
            ts[wave][r * 130 + e] = fmaxf(t, 0.0f);
        }
    }
    __syncthreads();

    int mn = lane & 15, koff = (lane >> 4) << 1;
    const float* tr = &ts[wave][mn * 130];
    v8f acc = {};
    for (int k = 0; k < Ev; k += 4) {
        v2f a, bb;
        a[0]  = tr[k + koff];
        a[1]  = tr[k + koff + 1];
        bb[0] = w2p[(k + koff) * 16 + mn];
        bb[1] = w2p[(k + koff + 1) * 16 + mn];
        acc = wmma4(a, bb, acc);
    }

    if (mn < 8) {                                // N index = head, only 8 valid
        float pb = pb2[mn];
        int rbase = (lane >> 4) << 3;
#pragma unroll
        for (int rr = 0; rr < 8; ++rr) {
            int j = j0 + rbase + rr;
            bias[(((size_t)(b * Hv + mn) * Nv + i) * Nv) + j] = acc[rr] + pb;
        }
    }
}

// ---------------------------------------------------------------------------
// Fused attention: one wave32 per (b,h,i). QK^T*scale + bias, mask,
// softmax in registers (12 keys/lane), AV with cross-lane reduce.
// ---------------------------------------------------------------------------
__global__ void attn_kernel(const float* __restrict__ qkv, const float* __restrict__ bias,
                            const unsigned char* __restrict__ mask, float* __restrict__ out) {
    int gw = (blockIdx.x * blockDim.x + threadIdx.x) >> 5;
    int lane = threadIdx.x & 31;
    if (gw >= Bv * Hv * Nv) return;
    int i = gw % Nv;
    int bh = gw / Nv;
    int h = bh % Hv;
    int b = bh / Hv;

    const float* qp = qkv + ((size_t)(b * Nv + i) * (3 * Ev)) + h * HD;
    float q[HD];
#pragma unroll
    for (int d = 0; d < HD; ++d) q[d] = qp[d];

    const float* biasrow = bias + ((size_t)(b * Hv + h) * Nv + i) * Nv;
    float s[12];
#pragma unroll
    for (int c = 0; c < 12; ++c) {
        int j = lane + 32 * c;
        const float* kp = qkv + ((size_t)(b * Nv + j) * (3 * Ev)) + Ev + h * HD;
        float dot = 0.0f;
#pragma unroll
        for (int d = 0; d < HD; ++d) dot = fmaf(q[d], kp[d], dot);
        float lg = dot * 0.25f + biasrow[j];     // scale = hd^-0.5 = 0.25
        s[c] = mask[b * Nv + j] ? lg : -1e9f;
    }
    float mx = -3.0e38f;
#pragma unroll
    for (int c = 0; c < 12; ++c) mx = fmaxf(mx, s[c]);
#pragma unroll
    for (int o = 16; o >= 1; o >>= 1) mx = fmaxf(mx, __shfl_xor(mx, o));
    float sum = 0.0f;
#pragma unroll
    for (int c = 0; c < 12; ++c) { s[c] = expf(s[c] - mx); sum += s[c]; }
#pragma unroll
    for (int o = 16; o >= 1; o >>= 1) sum += __shfl_xor(sum, o);
    float inv = 1.0f / sum;

    float acc[HD];
#pragma unroll
    for (int d = 0; d < HD; ++d) acc[d] = 0.0f;
#pragma unroll
    for (int c = 0; c < 12; ++c) {
        int j = lane + 32 * c;
        const float* vp = qkv + ((size_t)(b * Nv + j) * (3 * Ev)) + 2 * Ev + h * HD;
#pragma unroll
        for (int d = 0; d < HD; ++d) acc[d] = fmaf(s[c], vp[d], acc[d]);
    }
#pragma unroll
    for (int d = 0; d < HD; ++d) {
        float v = acc[d];
#pragma unroll
        for (int o = 16; o >= 1; o >>= 1) v += __shfl_xor(v, o);
        if (lane == 0) out[(size_t)(b * Nv + i) * Ev + h * HD + d] = v * inv;
    }
}

// ---------------------------------------------------------------------------
// Masked max/mean pool -> first 128 of concat buffer o[b, 0:128]
// ---------------------------------------------------------------------------
__global__ void pool_kernel(const float* __restrict__ feat, const unsigned char* __restrict__ mask,
                            float* __restrict__ o) {
    int b = blockIdx.x;
    int e = threadIdx.x;           // 128
    float mx = -1e9f, sum = 0.0f, cnt = 0.0f;
    for (int n = 0; n < Nv; ++n) {
        float f = feat[(size_t)(b * Nv + n) * Ev + e];
        if (mask[b * Nv + n]) { mx = fmaxf(mx, f); sum += f; cnt += 1.0f; }
    }
    float mean = sum / (cnt + 1e-6f);
    o[(size_t)b * 256 + e] = (cnt == 0.0f) ? mean : mx;
}

// ---------------------------------------------------------------------------
// Param feature path -> o[b, 128:256]
// ---------------------------------------------------------------------------
__global__ void pfeat_kernel(const float* __restrict__ params,
                             const float* __restrict__ w1, const float* __restrict__ b1,
                             const float* __restrict__ g, const float* __restrict__ bn,
                             const float* __restrict__ w2, const float* __restrict__ b2,
                             float* __restrict__ o) {
    int b = blockIdx.x;
    int e = threadIdx.x;           // 128
    float t = b1[e];
#pragma unroll
    for (int c = 0; c < 5; ++c) t = fmaf(params[b * 5 + c], w1[c * Ev + e], t);
    t = gelu_exact(t);
    __shared__ float s1[4], s2[4];
    float sum = t, sq = t * t;
#pragma unroll
    for (int of = 16; of >= 1; of >>= 1) {
        sum += __shfl_xor(sum, of);
        sq  += __shfl_xor(sq, of);
    }
    if ((e & 31) == 0) { s1[e >> 5] = sum; s2[e >> 5] = sq; }
    __syncthreads();
    float m = (s1[0] + s1[1] + s1[2] + s1[3]) * (1.0f / Ev);
    float var = (s2[0] + s2[1] + s2[2] + s2[3]) * (1.0f / Ev) - m * m;
    float p = (t - m) * rsqrtf(var + 1e-5f) * g[e] + bn[e];
    __shared__ float ps[Ev];
    ps[e] = p;
    __syncthreads();
    float a = b2[e];
    for (int k = 0; k < Ev; ++k) a = fmaf(ps[k], w2[k * Ev + e], a);
    o[(size_t)b * 256 + Ev + e] = gelu_exact(a);
}

// ---------------------------------------------------------------------------
// Small classifier layer: out[b,j] = act(in[b,:]@w + bias)
// ---------------------------------------------------------------------------
__global__ void mlp_kernel(const float* __restrict__ in, const float* __restrict__ w,
                           const float* __restrict__ bias, float* __restrict__ out,
                           int K, int Nc, int act) {
    int b = blockIdx.x;
    int j = threadIdx.x;           // 256
    __shared__ float sh[256];
    for (int k = j; k < K; k += 256) sh[k] = in[(size_t)b * K + k];
    __syncthreads();
    if (j < Nc) {
        float a = bias[j];
        for (int k = 0; k < K; ++k) a = fmaf(sh[k], w[k * Nc + j], a);
        if (act) a = gelu_exact(a);
        out[(size_t)b * Nc + j] = a;
    }
}

// ---------------------------------------------------------------------------
// host launcher
// ---------------------------------------------------------------------------
static inline void launch_gemm(const float* A, const float* W, const float* bias,
                               const float* res, float* C, int M, int K, int Nc, int act,
                               hipStream_t stream) {
    int waves = (M / 16) * (Nc / 64);
    int grid = (waves * 32 + 255) / 256;
    wmma_gemm_kernel<<<grid, 256, 0, stream>>>(A, W, bias, res, C, M, K, Nc, act);
}

extern "C" void kernel_launch(void* const* d_in, const int* in_sizes, int n_in,
                              void* d_out, int out_size, void* d_ws, size_t ws_size,
                              hipStream_t stream) {
    (void)in_sizes; (void)n_in; (void)out_size; (void)ws_size;
    const float* hits = (const float*)d_in[0];
    const unsigned char* mask = (const unsigned char*)d_in[1];
    const float* params = (const float*)d_in[2];
    // pytree flatten order (dict keys sorted): blocks first (list of 4 dicts,
    // each sorted: fb1,fb2,fw1,fw2,ln1b,ln1g,ln2b,ln2g,out_b,out_w,pb1,pb2,pw1,pw2,qkv_w),
    // then c_b1,c_b2,c_b3,c_w1,c_w2,c_w3,emb_b,emb_bn,emb_g,emb_w,
    // pf_b1,pf_b2,pf_bn,pf_g,pf_w1,pf_w2,pos_b,pos_w
    auto BLK = [&](int l, int k) { return (const float*)d_in[3 + l * 15 + k]; };
    enum { FB1=0, FB2, FW1, FW2, LN1B, LN1G, LN2B, LN2G, OUT_B, OUT_W, PB1, PB2, PW1, PW2, QKV_W };
    int wb = 3 + 4 * 15;   // 63
    const float* c_b1  = (const float*)d_in[wb + 0];
    const float* c_b2  = (const float*)d_in[wb + 1];
    const float* c_b3  = (const float*)d_in[wb + 2];
    const float* c_w1  = (const float*)d_in[wb + 3];
    const float* c_w2  = (const float*)d_in[wb + 4];
    const float* c_w3  = (const float*)d_in[wb + 5];
    const float* emb_b = (const float*)d_in[wb + 6];
    const float* emb_bn= (const float*)d_in[wb + 7];
    const float* emb_g = (const float*)d_in[wb + 8];
    const float* emb_w = (const float*)d_in[wb + 9];
    const float* pf_b1 = (const float*)d_in[wb + 10];
    const float* pf_b2 = (const float*)d_in[wb + 11];
    const float* pf_bn = (const float*)d_in[wb + 12];
    const float* pf_g  = (const float*)d_in[wb + 13];
    const float* pf_w1 = (const float*)d_in[wb + 14];
    const float* pf_w2 = (const float*)d_in[wb + 15];
    const float* pos_b = (const float*)d_in[wb + 16];
    const float* pos_w = (const float*)d_in[wb + 17];

    const int M = Bv * Nv;                 // 3072 rows
    float* W = (float*)d_ws;
    float* feat    = W;                    size_t off = (size_t)M * Ev;
    float* xbuf    = W + off;              off += (size_t)M * Ev;
    float* tmp     = W + off;              off += (size_t)M * Ev;
    float* qkv     = W + off;              off += (size_t)M * 3 * Ev;
    float* attn_o  = W + off;              off += (size_t)M * Ev;
    float* ffn_h   = W + off;              off += (size_t)M * 4 * Ev;
    float* biasbuf = W + off;              off += (size_t)Bv * Hv * Nv * Nv; // 9.4M
    float* denom   = W + off;              off += 32;
    float* catbuf  = W + off;              off += (size_t)Bv * 256;
    float* c1      = W + off;              off += (size_t)Bv * 256;
    float* c2      = W + off;              off += (size_t)Bv * 128;

    // embedding + pair normalization denominators
    embed_kernel<<<M, Ev, 0, stream>>>(hits, emb_w, emb_b, emb_g, emb_bn, pos_w, pos_b, feat);
    pair_denom_kernel<<<Bv * 3, 256, 0, stream>>>(hits, denom);

    const int pb_blocks = (Bv * Nv * Nv) / (16 * PB_WAVES);   // 18432, exact
    for (int l = 0; l < 4; ++l) {
        // qkv = feat @ qkv_w (no bias)
        launch_gemm(feat, BLK(l, QKV_W), nullptr, nullptr, qkv, M, Ev, 3 * Ev, 0, stream);
        // pair bias for this layer (WMMA stage 2)
        pair_bias_kernel<<<pb_blocks, 32 * PB_WAVES, 0, stream>>>(
            hits, denom, BLK(l, PW1), BLK(l, PB1), BLK(l, PW2), BLK(l, PB2), biasbuf);
        // fused attention
        attn_kernel<<<(Bv * Hv * Nv * 32 + 255) / 256, 256, 0, stream>>>(qkv, biasbuf, mask, attn_o);
        // out projection + residual, then LN1 -> x
        launch_gemm(attn_o, BLK(l, OUT_W), BLK(l, OUT_B), feat, tmp, M, Ev, Ev, 0, stream);
        ln_kernel<<<M, Ev, 0, stream>>>(tmp, xbuf, BLK(l, LN1G), BLK(l, LN1B));
        // FFN
        launch_gemm(xbuf, BLK(l, FW1), BLK(l, FB1), nullptr, ffn_h, M, Ev, 4 * Ev, 1, stream);
        launch_gemm(ffn_h, BLK(l, FW2), BLK(l, FB2), xbuf, tmp, M, 4 * Ev, Ev, 0, stream);
        ln_kernel<<<M, Ev, 0, stream>>>(tmp, feat, BLK(l, LN2G), BLK(l, LN2B));
    }

    // pooling + param path -> concat(256)
    pool_kernel<<<Bv, Ev, 0, stream>>>(feat, mask, catbuf);
    pfeat_kernel<<<Bv, Ev, 0, stream>>>(params, pf_w1, pf_b1, pf_g, pf_bn, pf_w2, pf_b2, catbuf);

    // classifier head
    mlp_kernel<<<Bv, 256, 0, stream>>>(catbuf, c_w1, c_b1, c1, 256, 256, 1);
    mlp_kernel<<<Bv, 256, 0, stream>>>(c1, c_w2, c_b2, c2, 256, 128, 1);
    mlp_kernel<<<Bv, 256, 0, stream>>>(c2, c_w3, c_b3, (float*)d_out, 128, 2, 0);
}